// AttentionModule_79319456022824
// MI455X (gfx1250) — compile-verified
//
#include <hip/hip_runtime.h>
#include <math.h>

typedef float v2f __attribute__((ext_vector_type(2)));
typedef float v8f __attribute__((ext_vector_type(8)));
typedef int   b128_t __attribute__((vector_size(16)));   // int32x4 payload type for async b128

#define N_NODES   1000000
#define N_GRAPHS  16384
#define NF        128   // node features
#define GF        64    // global features
#define KTOT      192   // NF + GF
#define HID       64

// CDNA5 async global->LDS path (guarded: falls back to float4 copies if the
// toolchain / host pass doesn't declare the builtins)
#if __has_builtin(__builtin_amdgcn_global_load_async_to_lds_b128) && \
    __has_builtin(__builtin_amdgcn_s_wait_asynccnt)
#define USE_ASYNC_LDS 1
typedef __attribute__((address_space(1))) b128_t* global_b128_ptr;
typedef __attribute__((address_space(3))) void*   lds_vptr;
typedef __attribute__((address_space(3))) b128_t* lds_b128_ptr;
#define ASYNC_B128(srcp, dstp) \
    __builtin_amdgcn_global_load_async_to_lds_b128( \
        (global_b128_ptr)(void*)(srcp), (lds_b128_ptr)(lds_vptr)(dstp), 0, 0)
#else
#define USE_ASYNC_LDS 0
#endif

// ---------------- Kernel 1: per-node MLP score via f32 WMMA ----------------
#define K1_THREADS      256
#define NODES_PER_BLOCK 128
#define XU_STRIDE       196   // floats; 784B row: 16B aligned, bank-conflict-free b64 frags
#define W1P_STRIDE      66    // float2 per k-pair row (64 cols + pad)

__global__ __launch_bounds__(K1_THREADS)
void score_kernel(const float* __restrict__ x,
                  const float* __restrict__ u,
                  const int*   __restrict__ batch,
                  const float* __restrict__ W1,
                  const float* __restrict__ b1,
                  const float* __restrict__ W2,
                  const float* __restrict__ b2,
                  float* __restrict__ sOut)
{
    __shared__ float xu[NODES_PER_BLOCK * XU_STRIDE];            // ~98 KB
    __shared__ v2f   w1p[(KTOT / 2) * W1P_STRIDE];               // ~50 KB
    __shared__ float b1s[HID];
    __shared__ float w2s[HID];

    const int tid  = threadIdx.x;
    const int base = blockIdx.x * NODES_PER_BLOCK;

    // ---- stage x -> xu[:, 0:128] (coalesced b128; async direct-to-LDS on CDNA5) ----
    // OOB node rows are clamped to a valid row: their scores are computed but
    // never stored, so no zero-fill is required and EXEC stays uniform.
    for (int idx = tid; idx < NODES_PER_BLOCK * (NF / 4); idx += K1_THREADS) {
        int i = idx >> 5;            // 32 float4 chunks per row
        int c = idx & 31;
        int g = base + i;
        if (g >= N_NODES) g = N_NODES - 1;
        const float* src = x + (size_t)g * NF + c * 4;
        float*       dst = &xu[i * XU_STRIDE + c * 4];
#if USE_ASYNC_LDS
        ASYNC_B128(src, dst);
#else
        *reinterpret_cast<float4*>(dst) = *reinterpret_cast<const float4*>(src);
#endif
    }
    // ---- stage u[batch[node]] -> xu[:, 128:192] (gather; sorted batch => L2 reuse) ----
    for (int idx = tid; idx < NODES_PER_BLOCK * (GF / 4); idx += K1_THREADS) {
        int i = idx >> 4;            // 16 float4 chunks per row
        int c = idx & 15;
        int g = base + i;
        if (g >= N_NODES) g = N_NODES - 1;
        int gr = batch[g];
        const float* src = u + (size_t)gr * GF + c * 4;
        float*       dst = &xu[i * XU_STRIDE + NF + c * 4];
#if USE_ASYNC_LDS
        ASYNC_B128(src, dst);
#else
        *reinterpret_cast<float4*>(dst) = *reinterpret_cast<const float4*>(src);
#endif
    }
    // ---- stage W1 as interleaved k-pairs: w1p[k2][c] = (W1[2k2][c], W1[2k2+1][c]) ----
    for (int idx = tid; idx < (KTOT / 2) * HID; idx += K1_THREADS) {
        int k2 = idx >> 6;
        int c  = idx & 63;
        v2f v;
        v.x = W1[(2 * k2    ) * HID + c];
        v.y = W1[(2 * k2 + 1) * HID + c];
        w1p[k2 * W1P_STRIDE + c] = v;
    }
    if (tid < HID) { b1s[tid] = b1[tid]; w2s[tid] = W2[tid]; }

#if USE_ASYNC_LDS
    __builtin_amdgcn_s_wait_asynccnt(0);
#endif
    __syncthreads();

    const int lane = tid & 31;
    const int wave = tid >> 5;          // 8 waves -> 8 x 16-node tiles
    const int m    = lane & 15;
    const int half = lane >> 4;

    // A fragment source: node row (wave*16 + m), K pair starting at 2*half
    const float* arow = &xu[(wave * 16 + m) * XU_STRIDE + 2 * half];

    v8f acc0 = {}, acc1 = {}, acc2 = {}, acc3 = {};

    #pragma unroll 4
    for (int k0 = 0; k0 < KTOT; k0 += 4) {
        v2f a = *reinterpret_cast<const v2f*>(arow + k0);
        const v2f* brow = &w1p[((k0 >> 1) + half) * W1P_STRIDE + m];
        v2f bf0 = brow[0];
        v2f bf1 = brow[16];
        v2f bf2 = brow[32];
        v2f bf3 = brow[48];
        acc0 = __builtin_amdgcn_wmma_f32_16x16x4_f32(false, a, false, bf0, (short)0, acc0, false, false);
        acc1 = __builtin_amdgcn_wmma_f32_16x16x4_f32(false, a, false, bf1, (short)0, acc1, false, false);
        acc2 = __builtin_amdgcn_wmma_f32_16x16x4_f32(false, a, false, bf2, (short)0, acc2, false, false);
        acc3 = __builtin_amdgcn_wmma_f32_16x16x4_f32(false, a, false, bf3, (short)0, acc3, false, false);
    }

    // ---- epilogue: +b1, relu, dot with W2, 16-lane reduce, store score ----
    const float b2v = b2[0];
    const float w2a = w2s[m], w2b = w2s[16 + m], w2c = w2s[32 + m], w2d = w2s[48 + m];
    const float b1a = b1s[m], b1b = b1s[16 + m], b1c = b1s[32 + m], b1d = b1s[48 + m];
    #pragma unroll
    for (int r = 0; r < 8; ++r) {
        float h0 = fmaxf(acc0[r] + b1a, 0.f);
        float h1 = fmaxf(acc1[r] + b1b, 0.f);
        float h2 = fmaxf(acc2[r] + b1c, 0.f);
        float h3 = fmaxf(acc3[r] + b1d, 0.f);
        float p = fmaf(h0, w2a, fmaf(h1, w2b, fmaf(h2, w2c, h3 * w2d)));
        p += __shfl_xor(p, 1);
        p += __shfl_xor(p, 2);
        p += __shfl_xor(p, 4);
        p += __shfl_xor(p, 8);
        if (m == 0) {
            int node = base + wave * 16 + r + 8 * half;   // C layout: VGPR r -> M = r + 8*half
            if (node < N_NODES) sOut[node] = p + b2v;
        }
    }
}

// -------- Kernel 2: per-graph stable softmax + attention-weighted pool --------
__global__ __launch_bounds__(128)
void softmax_pool_kernel(const float* __restrict__ x,
                         const int*   __restrict__ batch,
                         const float* __restrict__ sArr,
                         float* __restrict__ pooled,
                         float* __restrict__ attn)
{
    __shared__ float red[128];
    __shared__ int   bounds[2];
    const int g   = blockIdx.x;
    const int tid = threadIdx.x;

    if (tid < 2) {   // lower_bound(batch, g) and lower_bound(batch, g+1): sorted batch
        int target = g + tid;
        int lo = 0, hi = N_NODES;
        while (lo < hi) {
            int mid = (lo + hi) >> 1;
            if (batch[mid] < target) lo = mid + 1; else hi = mid;
        }
        bounds[tid] = lo;
    }
    __syncthreads();
    const int beg = bounds[0], end = bounds[1];

    // segment max
    float mloc = -INFINITY;
    for (int n = beg + tid; n < end; n += 128) mloc = fmaxf(mloc, sArr[n]);
    red[tid] = mloc; __syncthreads();
    for (int off = 64; off > 0; off >>= 1) {
        if (tid < off) red[tid] = fmaxf(red[tid], red[tid + off]);
        __syncthreads();
    }
    const float mval = red[0];
    __syncthreads();

    // segment sum of exp(s - m)
    float zloc = 0.f;
    for (int n = beg + tid; n < end; n += 128) zloc += expf(sArr[n] - mval);
    red[tid] = zloc; __syncthreads();
    for (int off = 64; off > 0; off >>= 1) {
        if (tid < off) red[tid] += red[tid + off];
        __syncthreads();
    }
    const float z = red[0];
    const float invz = (end > beg) ? (1.f / z) : 0.f;

    // attn output
    for (int n = beg + tid; n < end; n += 128) attn[n] = expf(sArr[n] - mval) * invz;

    // attention-weighted pooling: thread tid = feature column
    float accp = 0.f;
    for (int n = beg; n < end; ++n) {
        if (n + 8 < end) __builtin_prefetch(x + (size_t)(n + 8) * NF + tid, 0, 0);
        float a = expf(sArr[n] - mval) * invz;   // broadcast load of sArr[n], cheap exp
        accp = fmaf(x[(size_t)n * NF + tid], a, accp);
    }
    pooled[(size_t)g * NF + tid] = accp;
}

// ------------------------------- launcher -------------------------------
extern "C" void kernel_launch(void* const* d_in, const int* in_sizes, int n_in,
                              void* d_out, int out_size, void* d_ws, size_t ws_size,
                              hipStream_t stream)
{
    const float* x     = (const float*)d_in[0];
    const float* u     = (const float*)d_in[1];
    const int*   batch = (const int*)  d_in[2];
    const float* W1    = (const float*)d_in[3];
    const float* b1    = (const float*)d_in[4];
    const float* W2    = (const float*)d_in[5];
    const float* b2    = (const float*)d_in[6];

    float* out    = (float*)d_out;
    float* pooled = out;                              // [G, 128]
    float* attn   = out + (size_t)N_GRAPHS * NF;      // [N]
    float* sArr   = (float*)d_ws;                     // [N] scores scratch

    dim3 g1((N_NODES + NODES_PER_BLOCK - 1) / NODES_PER_BLOCK);
    score_kernel<<<g1, K1_THREADS, 0, stream>>>(x, u, batch, W1, b1, W2, b2, sArr);
    softmax_pool_kernel<<<N_GRAPHS, 128, 0, stream>>>(x, batch, sArr, pooled, attn);
}